// EBP_binaryNet_40003325394948
// MI455X (gfx1250) — compile-verified
//
#include <hip/hip_runtime.h>
#include <hip/hip_bf16.h>
#include <math.h>

typedef __attribute__((ext_vector_type(2))) float v2f;
typedef __attribute__((ext_vector_type(4))) float v4f;
typedef __attribute__((ext_vector_type(8))) float v8f;

#define SQ2PI 0.79788456f
#define Hh 512
#define Mb 512
#define DIN 784

// ---------------- elementwise: m = 2*sigmoid(w)-1 = tanh(w/2) ----------------
__global__ void map_m_kernel(const float* __restrict__ w, float* __restrict__ m, int n) {
    int i = blockIdx.x * blockDim.x + threadIdx.x;
    int stride = gridDim.x * blockDim.x;
    for (; i < n; i += stride) m[i] = tanhf(0.5f * w[i]);
}

// ---------------- csum[j] = sum_i (1 - m[i][j]^2) ----------------
__global__ void colsum_kernel(const float* __restrict__ m, int rows, int cols,
                              float* __restrict__ out) {
    int j = blockIdx.x * blockDim.x + threadIdx.x;
    if (j >= cols) return;
    float s = 0.0f;
    for (int i = 0; i < rows; ++i) {
        float v = m[(size_t)i * cols + j];
        s += 1.0f - v * v;
    }
    out[j] = s;
}

// ---------------- fused EBP layer: dual WMMA GEMM + tanh epilogue ----------------
// h = Ax @ B   (M=512, N=512, K)          via V_WMMA_F32_16X16X4_F32
// s = Ad @ B^2 (DUAL only; B^2 squared in registers, no extra memory)
// xout = tanh(SQ2PI*(h+th)/sqrt(s+csum)); dout = 1 - xout^2
// One wave per 16x16 C tile; 1024 tiles; 8 waves per 256-thread block -> 128 blocks.
// Template specialization removes all inner-loop control flow; one-stage
// software pipeline overlaps L2 loads with WMMA issue.
template <bool DUAL>
__global__ void __launch_bounds__(256)
ebp_layer_kernel(const float* __restrict__ Ax, int lda,
                 const float* __restrict__ Ad,        // ld = 512 (DUAL only)
                 const float* __restrict__ B,         // K x 512 row-major
                 int K,
                 const float* __restrict__ csum,      // [512]
                 const float* __restrict__ th,        // [512]
                 float* __restrict__ xout,            // 512 x 512
                 float* __restrict__ dout)            // 512 x 512
{
    const int wave  = (blockIdx.x * blockDim.x + threadIdx.x) >> 5;
    const int lane  = threadIdx.x & 31;
    const int half  = lane >> 4;        // 0: K{0,1}, 1: K{2,3} within the 4-step
    const int l15   = lane & 15;
    const int tM    = wave >> 5;        // 32 tiles along M
    const int tN    = wave & 31;        // 32 tiles along N

    const int rowA = tM * 16 + l15;     // A-frag row for this lane
    const int colB = tN * 16 + l15;     // B-frag column for this lane

    const float* pA1 = Ax + (size_t)rowA * lda + 2 * half;           // +kk
    const float* pA2 = DUAL ? (Ad + (size_t)rowA * Hh + 2 * half) : nullptr;
    const float* pB  = B + (size_t)(2 * half) * Hh + colB;           // +kk*Hh

    v8f acc_h = {};
    v8f acc_s = {};

    // ---- software-pipelined K loop (K % 4 == 0, K >= 8) ----
    v2f a1, a2, b;
    a1.x = pA1[0]; a1.y = pA1[1];
    b.x  = pB[0];  b.y  = pB[Hh];
    if (DUAL) { a2.x = pA2[0]; a2.y = pA2[1]; }

    for (int kk = 0; kk < K - 4; kk += 4) {
        // prefetch next fragments (kk+4) before consuming current ones
        v2f a1n, a2n, bn;
        a1n.x = pA1[kk + 4]; a1n.y = pA1[kk + 5];
        bn.x  = pB[(size_t)(kk + 4) * Hh];
        bn.y  = pB[(size_t)(kk + 5) * Hh];
        if (DUAL) { a2n.x = pA2[kk + 4]; a2n.y = pA2[kk + 5]; }

        acc_h = __builtin_amdgcn_wmma_f32_16x16x4_f32(
            false, a1, false, b, (short)0, acc_h, false, false);
        if (DUAL) {
            v2f b2; b2.x = b.x * b.x; b2.y = b.y * b.y;
            acc_s = __builtin_amdgcn_wmma_f32_16x16x4_f32(
                false, a2, false, b2, (short)0, acc_s, false, false);
        }
        a1 = a1n; b = bn;
        if (DUAL) a2 = a2n;
    }
    acc_h = __builtin_amdgcn_wmma_f32_16x16x4_f32(
        false, a1, false, b, (short)0, acc_h, false, false);
    if (DUAL) {
        v2f b2; b2.x = b.x * b.x; b2.y = b.y * b.y;
        acc_s = __builtin_amdgcn_wmma_f32_16x16x4_f32(
            false, a2, false, b2, (short)0, acc_s, false, false);
    }

    const float cs = csum[colB];
    const float tc = th[colB];
#pragma unroll
    for (int v = 0; v < 8; ++v) {
        const int row = tM * 16 + v + 8 * half;   // C layout: VGPR v, lane-half
        const float h = acc_h[v] + tc;
        const float s = (DUAL ? acc_s[v] : 0.0f) + cs;
        const float xn = tanhf(SQ2PI * h * rsqrtf(s));
        const size_t idx = (size_t)row * Hh + colB;
        xout[idx] = xn;
        dout[idx] = 1.0f - xn * xn;
    }
}

// ---------------- last layer: per-row dot with mlast / mlast^2 ----------------
__global__ void __launch_bounds__(256)
last_layer_kernel(const float* __restrict__ x4, const float* __restrict__ d4,
                  const float* __restrict__ mlast, const float* __restrict__ csumlast,
                  const float* __restrict__ thlast, const int* __restrict__ target,
                  float* __restrict__ out_h, float* __restrict__ out_logp,
                  float* __restrict__ lossterm, float* __restrict__ abserr)
{
    const int r    = (blockIdx.x * blockDim.x + threadIdx.x) >> 5;   // 512 waves
    const int lane = threadIdx.x & 31;
    float h = 0.0f, s = 0.0f;
    for (int j = lane; j < Hh; j += 32) {
        const float ml = mlast[j];
        h += x4[(size_t)r * Hh + j] * ml;
        s += d4[(size_t)r * Hh + j] * ml * ml;
    }
#pragma unroll
    for (int off = 16; off > 0; off >>= 1) {
        h += __shfl_xor(h, off, 32);
        s += __shfl_xor(s, off, 32);
    }
    if (lane == 0) {
        const float hb   = h + thlast[0];
        const float dsig = s + csumlast[0];
        const float hl   = SQ2PI * hb * rsqrtf(dsig);
        // stable log_sigmoid
        const float lp = fminf(hl, 0.0f)  - log1pf(expf(-fabsf(hl)));
        const float lq = fminf(-hl, 0.0f) - log1pf(expf(-fabsf(hl)));
        const float y  = (float)target[r];
        out_h[r]    = hb;
        out_logp[r] = lp;
        lossterm[r] = -(y * lp + (1.0f - y) * lq);
        const float pred = (hl > 0.0f) ? 1.0f : 0.0f;   // sigmoid(hl) > 0.5
        abserr[r] = fabsf(pred - y);
    }
}

// ---------------- final reduction: loss + fraction_correct ----------------
__global__ void finalize_kernel(const float* __restrict__ lossterm,
                                const float* __restrict__ abserr,
                                float* __restrict__ out_loss,
                                float* __restrict__ out_frac)
{
    __shared__ float sl[Mb];
    __shared__ float sa[Mb];
    const int t = threadIdx.x;
    sl[t] = lossterm[t];
    sa[t] = abserr[t];
    __syncthreads();
    for (int off = Mb / 2; off > 0; off >>= 1) {
        if (t < off) { sl[t] += sl[t + off]; sa[t] += sa[t + off]; }
        __syncthreads();
    }
    if (t == 0) {
        *out_loss = sl[0] / (float)Mb;
        *out_frac = ((float)Mb - sa[0]) / (float)Mb;
    }
}

// ---------------- xcov zero-fill: non-temporal b128 streaming stores ----------------
__global__ void fill_zero16_kernel(v4f* __restrict__ p, size_t n4) {
    size_t i = (size_t)blockIdx.x * blockDim.x + threadIdx.x;
    const size_t stride = (size_t)gridDim.x * blockDim.x;
    const v4f z = {0.0f, 0.0f, 0.0f, 0.0f};
    for (; i < n4; i += stride)
        __builtin_nontemporal_store(z, &p[i]);   // NT hint: don't occupy L2
}

__global__ void fill_diag_kernel(float* __restrict__ xcov, const float* __restrict__ d4) {
    const int t = blockIdx.x * blockDim.x + threadIdx.x;   // 0 .. 512*512-1
    const int b = t >> 9;
    const int i = t & 511;
    xcov[(size_t)b * (Hh * Hh) + (size_t)i * (Hh + 1)] = d4[t];
}

// ---------------- host-side orchestration ----------------
extern "C" void kernel_launch(void* const* d_in, const int* in_sizes, int n_in,
                              void* d_out, int out_size, void* d_ws, size_t ws_size,
                              hipStream_t stream) {
    (void)in_sizes; (void)n_in; (void)out_size; (void)ws_size;

    const float* x      = (const float*)d_in[0];
    const float* w0     = (const float*)d_in[1];
    const float* w1     = (const float*)d_in[2];
    const float* w2     = (const float*)d_in[3];
    const float* w3     = (const float*)d_in[4];
    /* w4 (d_in[5]) unused by the reference */
    const float* wlast  = (const float*)d_in[6];
    const float* th0    = (const float*)d_in[7];
    const float* th1    = (const float*)d_in[8];
    const float* th2    = (const float*)d_in[9];
    const float* th3    = (const float*)d_in[10];
    /* th4 (d_in[11]) unused */
    const float* thlast = (const float*)d_in[12];
    const int*   target = (const int*)d_in[13];

    float* ws = (float*)d_ws;
    // workspace layout (floats)
    float* m0       = ws;                      // 784*512
    float* m1       = m0  + (size_t)DIN * Hh;  // 512*512
    float* m2       = m1  + (size_t)Hh * Hh;
    float* m3       = m2  + (size_t)Hh * Hh;
    float* mlast    = m3  + (size_t)Hh * Hh;   // 512
    float* csum0    = mlast + Hh;              // 512 each
    float* csum1    = csum0 + Hh;
    float* csum2    = csum1 + Hh;
    float* csum3    = csum2 + Hh;
    float* csumlast = csum3 + Hh;              // 1 (padded to 64)
    float* x1       = csumlast + 64;           // 512*512 each
    float* d1       = x1 + (size_t)Hh * Hh;
    float* x2       = d1 + (size_t)Hh * Hh;
    float* d2       = x2 + (size_t)Hh * Hh;
    float* x3       = d2 + (size_t)Hh * Hh;
    float* d3       = x3 + (size_t)Hh * Hh;
    float* x4       = d3 + (size_t)Hh * Hh;
    float* d4       = x4 + (size_t)Hh * Hh;
    float* lossterm = d4 + (size_t)Hh * Hh;    // 512
    float* abserr   = lossterm + Hh;           // 512

    // output layout (floats): hlastbar[512] | logp[512] | xcov[512^3] | loss | frac
    float* out       = (float*)d_out;
    float* out_h     = out;
    float* out_logp  = out + Mb;
    float* out_xcov  = out + 2 * Mb;
    const size_t N3  = (size_t)Hh * Hh * Hh;
    float* out_loss  = out + 2 * Mb + N3;
    float* out_frac  = out_loss + 1;

    // 1) m = tanh(w/2)
    map_m_kernel<<<512, 256, 0, stream>>>(w0, m0, DIN * Hh);
    map_m_kernel<<<512, 256, 0, stream>>>(w1, m1, Hh * Hh);
    map_m_kernel<<<512, 256, 0, stream>>>(w2, m2, Hh * Hh);
    map_m_kernel<<<512, 256, 0, stream>>>(w3, m3, Hh * Hh);
    map_m_kernel<<<2, 256, 0, stream>>>(wlast, mlast, Hh);

    // 2) column sums of (1 - m^2)
    colsum_kernel<<<2, 256, 0, stream>>>(m0, DIN, Hh, csum0);
    colsum_kernel<<<2, 256, 0, stream>>>(m1, Hh, Hh, csum1);
    colsum_kernel<<<2, 256, 0, stream>>>(m2, Hh, Hh, csum2);
    colsum_kernel<<<2, 256, 0, stream>>>(m3, Hh, Hh, csum3);
    colsum_kernel<<<1, 32, 0, stream>>>(mlast, Hh, 1, csumlast);

    // 3) layers (WMMA f32 16x16x4). Layer 3 intentionally reuses d2 (reference quirk).
    ebp_layer_kernel<false><<<128, 256, 0, stream>>>(x,  DIN, nullptr, m0, DIN, csum0, th0, x1, d1);
    ebp_layer_kernel<true ><<<128, 256, 0, stream>>>(x1, Hh,  d1,      m1, Hh,  csum1, th1, x2, d2);
    ebp_layer_kernel<true ><<<128, 256, 0, stream>>>(x2, Hh,  d2,      m2, Hh,  csum2, th2, x3, d3);
    ebp_layer_kernel<true ><<<128, 256, 0, stream>>>(x3, Hh,  d2,      m3, Hh,  csum3, th3, x4, d4);

    // 4) last layer + scalars
    last_layer_kernel<<<64, 256, 0, stream>>>(x4, d4, mlast, csumlast, thlast, target,
                                              out_h, out_logp, lossterm, abserr);
    finalize_kernel<<<1, Mb, 0, stream>>>(lossterm, abserr, out_loss, out_frac);

    // 5) xcov_4: 512 MB zero-fill with NT b128 stores, then diagonal scatter
    fill_zero16_kernel<<<4096, 256, 0, stream>>>((v4f*)out_xcov, N3 / 4);
    fill_diag_kernel<<<1024, 256, 0, stream>>>(out_xcov, d4);
}